// RddTransformer_61581241090557
// MI455X (gfx1250) — compile-verified
//
#include <hip/hip_runtime.h>
#include <math.h>

// Problem constants (match reference).
#define Bq  8
#define Nq  16384
#define Dq  768
#define Cq  8
#define NCq 2

typedef __attribute__((ext_vector_type(2))) float v2f;
typedef __attribute__((ext_vector_type(8))) float v8f;

#define CHUNK    512                  // instance rows per block
#define TROWS    8                    // rows per LDS tile (two K=4 WMMA groups)
#define TILES    (CHUNK / TROWS)      // 64 tile iterations per block
#define NWAVES   12                   // one wave per 64-dim group: 12*64 = 768
#define BLOCKT   (NWAVES * 32)        // 384 threads
#define LDS_ROW  776                  // padded row stride (floats); 776%64=8 -> conflict-free
#define BUF_FLOATS (TROWS * LDS_ROW)  // 6208 floats per buffer
#define BUF_BYTES  (BUF_FLOATS * 4)   // 24832 B per buffer
#define CHUNKS16 (Dq / 4)             // 192 16-byte chunks per row

// ---------------------------------------------------------------------------
// Zero the accumulation workspace (sums + counts) — required every launch.
// ---------------------------------------------------------------------------
__global__ void zero_ws(float* __restrict__ ws, int n) {
    int i = blockIdx.x * blockDim.x + threadIdx.x;
    if (i < n) ws[i] = 0.0f;
}

// ---------------------------------------------------------------------------
// Per-(bag, cluster) instance counts via LDS histogram then one global atomic.
// ---------------------------------------------------------------------------
__global__ void count_kernel(const int* __restrict__ labels,
                             float* __restrict__ counts) {
    __shared__ float scnt[Cq];
    const int b = blockIdx.x;
    if (threadIdx.x < Cq) scnt[threadIdx.x] = 0.0f;
    __syncthreads();
    for (int n = threadIdx.x; n < Nq; n += blockDim.x)
        atomicAdd(&scnt[labels[b * Nq + n]], 1.0f);
    __syncthreads();
    if (threadIdx.x < Cq)
        atomicAdd(&counts[b * Cq + threadIdx.x], scnt[threadIdx.x]);
}

// ---------------------------------------------------------------------------
// Segment-sum as onehot(labels) @ feat with f32 WMMA, fed by double-buffered
// async global->LDS B128 copies (ASYNCcnt pipeline, tile shared by 12 waves).
// grid = (Nq/CHUNK, Bq), block = 384 threads (12 waves).
//   wave w owns feature dims [w*64, w*64+63] (4 column tiles of 16).
//   A (16x4 f32): one-hot of 4 consecutive labels (rows 8..15 zero).
//   B (4x16 f32): 4 staged rows x 16 columns, read from LDS.
// ---------------------------------------------------------------------------
__global__ void __launch_bounds__(BLOCKT)
segsum_wmma(const float* __restrict__ feat,
            const int*   __restrict__ labels,
            float*       __restrict__ sums) {
    __shared__ float smem[2 * BUF_FLOATS];   // 49,664 B

    const int tid   = threadIdx.x;
    const int lane  = tid & 31;
    const int wave  = tid >> 5;          // 0..11 -> 64-dim group
    const int mrow  = lane & 15;
    const int khalf = lane >> 4;
    const int nBeg  = blockIdx.x * CHUNK;
    const int b     = blockIdx.y;

    const float* fb = feat + (size_t)b * Nq * Dq;
    const int*   lb = labels + b * Nq;
    const unsigned smem_base = (unsigned)(uintptr_t)(void*)&smem[0];

    // Precompute this lane's 4 async-copy chunk assignments (16 B each).
    // Block-wide: 12 waves * 4 * 32 lanes = 1536 chunks = 8 rows * 192 chunks.
    size_t   gofs[4];   // element offset within the tile's row block
    unsigned lofs[4];   // byte offset within one LDS buffer
#pragma unroll
    for (int i = 0; i < 4; ++i) {
        const int idx = wave * 128 + i * 32 + lane;
        const int row = idx / CHUNKS16;
        const int c4  = idx - row * CHUNKS16;
        gofs[i] = (size_t)row * Dq + c4 * 4;
        lofs[i] = (unsigned)(row * LDS_ROW + c4 * 4) * 4u;
    }

    v8f acc[4];
#pragma unroll
    for (int t = 0; t < 4; ++t) acc[t] = (v8f){};

    const int dcol = wave * 64 + mrow;

    // Prime the pipeline: tile 0 -> buffer 0.
    {
        const float* gb = fb + (size_t)nBeg * Dq;
#pragma unroll
        for (int i = 0; i < 4; ++i) {
            const float* gp = gb + gofs[i];
            const unsigned la = smem_base + lofs[i];
            asm volatile("global_load_async_to_lds_b128 %0, %1, off"
                         :: "v"(la), "v"(gp) : "memory");
        }
    }

    for (int t = 0; t < TILES; ++t) {
        const int nb = t & 1;
        if (t + 1 < TILES) {
            // Issue tile t+1 into the other buffer (its last readers finished
            // at iteration t-1, separated by that iteration's trailing barrier).
            const float* gb = fb + (size_t)(nBeg + (t + 1) * TROWS) * Dq;
            const unsigned lbase = smem_base + ((t + 1) & 1) * BUF_BYTES;
#pragma unroll
            for (int i = 0; i < 4; ++i) {
                const float* gp = gb + gofs[i];
                const unsigned la = lbase + lofs[i];
                asm volatile("global_load_async_to_lds_b128 %0, %1, off"
                             :: "v"(la), "v"(gp) : "memory");
            }
            asm volatile("s_wait_asynccnt 0x4" ::: "memory"); // tile t landed
        } else {
            asm volatile("s_wait_asynccnt 0x0" ::: "memory"); // drain
        }
        __syncthreads();   // all waves' portions of tile t are in LDS

        const float* sb = smem + nb * BUF_FLOATS;
        const int    n0 = nBeg + t * TROWS;
#pragma unroll
        for (int kg = 0; kg < 2; ++kg) {
            const int rl  = kg * 4 + khalf * 2;     // this lane's K rows (local)
            const int n0k = n0 + rl;
            const int l0  = lb[n0k];
            const int l1  = lb[n0k + 1];
            v2f a;
            a.x = (mrow == l0) ? 1.0f : 0.0f;       // labels in [0,8): rows 8..15 zero
            a.y = (mrow == l1) ? 1.0f : 0.0f;
            const float* s0 = sb + rl * LDS_ROW + dcol;
#pragma unroll
            for (int tt = 0; tt < 4; ++tt) {
                v2f bt;
                bt.x = s0[tt * 16];
                bt.y = s0[LDS_ROW + tt * 16];
                acc[tt] = __builtin_amdgcn_wmma_f32_16x16x4_f32(
                    false, a, false, bt, (short)0, acc[tt], false, false);
            }
        }
        __syncthreads();   // protect buffer nb before tile t+2 overwrites it
    }

    // C layout: VGPR j, lanes 0-15 => (M=j, N=lane). Rows 8..15 are zero.
    if (lane < 16) {
        const int dBase = wave * 64;
#pragma unroll
        for (int t = 0; t < 4; ++t) {
            float* dst = sums + (size_t)(b * Cq) * Dq + dBase + t * 16 + lane;
#pragma unroll
            for (int j = 0; j < Cq; ++j)
                atomicAdd(dst + (size_t)j * Dq, acc[t][j]);
        }
    }
}

// ---------------------------------------------------------------------------
// means -> head logits -> softmax -> score -> first-max argmax -> gather.
// ---------------------------------------------------------------------------
__global__ void finalize_kernel(const float* __restrict__ sums,
                                const float* __restrict__ counts,
                                const float* __restrict__ head_w,
                                const float* __restrict__ head_b,
                                float* __restrict__ out) {
    __shared__ float sl0[Bq * Cq], sl1[Bq * Cq], ssc[Bq * Cq];
    const int s = threadIdx.x;  // 0..63
    if (s < Bq * Cq) {
        const float inv = 1.0f / fmaxf(counts[s], 1.0f);
        float l0 = head_b[0], l1 = head_b[1];
        const float* sr = sums + (size_t)s * Dq;
        for (int d = 0; d < Dq; ++d) {
            const float m = sr[d] * inv;
            l0 = fmaf(m, head_w[d], l0);
            l1 = fmaf(m, head_w[Dq + d], l1);
        }
        const float mx = fmaxf(l0, l1);
        const float e0 = __expf(l0 - mx);
        const float e1 = __expf(l1 - mx);
        sl0[s] = l0;
        sl1[s] = l1;
        ssc[s] = 1.0f - e0 / (e0 + e1);  // 1 - probs[:, nor_index=0]
    }
    __syncthreads();
    if (s < Bq) {
        const int base = s * Cq;
        int   mi = 0;
        float bv = ssc[base];
        for (int c = 1; c < Cq; ++c) {   // strict > keeps first max (jnp.argmax)
            const float v = ssc[base + c];
            if (v > bv) { bv = v; mi = c; }
        }
        out[s * NCq + 0] = sl0[base + mi];
        out[s * NCq + 1] = sl1[base + mi];
    }
}

// ---------------------------------------------------------------------------
extern "C" void kernel_launch(void* const* d_in, const int* in_sizes, int n_in,
                              void* d_out, int out_size, void* d_ws, size_t ws_size,
                              hipStream_t stream) {
    const float* feat   = (const float*)d_in[0];
    const int*   labels = (const int*)  d_in[1];
    const float* head_w = (const float*)d_in[2];
    const float* head_b = (const float*)d_in[3];
    float*       out    = (float*)d_out;

    float* sums   = (float*)d_ws;           // [B*C, D]
    float* counts = sums + Bq * Cq * Dq;    // [B*C]

    const int nz = Bq * Cq * Dq + Bq * Cq;
    zero_ws<<<(nz + 255) / 256, 256, 0, stream>>>((float*)d_ws, nz);

    count_kernel<<<Bq, 256, 0, stream>>>(labels, counts);

    dim3 grid(Nq / CHUNK, Bq);              // 32 x 8 = 256 blocks, 12 waves each
    segsum_wmma<<<grid, BLOCKT, 0, stream>>>(feat, labels, sums);

    finalize_kernel<<<1, 64, 0, stream>>>(sums, counts, head_w, head_b, out);
}